// TranslationalEquivariantPooling3D_25391846654378
// MI455X (gfx1250) — compile-verified
//
#include <hip/hip_runtime.h>
#include <cstdint>

// ---------------------------------------------------------------------------
// Compile-time permutation table.
// _swap_stabilizers == gather by sigma, sigma[a*8+r] = bitrev3(r)*8 + bitrev3(PERM[a]).
// Z-update (cond bit) : q -> sigma[sigma[q]^m], m = 4/2/1 for axis 0/1/2
// X-update (cond bit) : q -> q^m,               m = 1/2/4 for axis 0/1/2
// Total map for class c (bits b0..b5 in application order Z0,X0,Z1,X1,Z2,X2):
//   T[c][q] = g0(g1(g2(g3(g4(g5(q))))))   (evaluate innermost g5 first)
// ---------------------------------------------------------------------------
struct TTab { unsigned char t[64][64]; };

constexpr int bitrev3_(int x) { return ((x & 1) << 2) | (x & 2) | ((x >> 2) & 1); }

constexpr TTab make_ttab() {
  TTab R{};
  int PERM[8] = {0, 6, 2, 4, 3, 5, 1, 7};
  int sig[64] = {};
  for (int a = 0; a < 8; a++)
    for (int r = 0; r < 8; r++)
      sig[a * 8 + r] = bitrev3_(r) * 8 + bitrev3_(PERM[a]);
  for (int c = 0; c < 64; c++) {
    for (int q = 0; q < 64; q++) {
      int idx = q;
      if (c & 32) idx ^= 4;                 // b5: X, axis 2
      if (c & 16) idx = sig[sig[idx] ^ 1];  // b4: Z, axis 2
      if (c & 8)  idx ^= 2;                 // b3: X, axis 1
      if (c & 4)  idx = sig[sig[idx] ^ 2];  // b2: Z, axis 1
      if (c & 2)  idx ^= 1;                 // b1: X, axis 0
      if (c & 1)  idx = sig[sig[idx] ^ 4];  // b0: Z, axis 0
      R.t[c][q] = (unsigned char)idx;
    }
  }
  return R;
}

__constant__ TTab g_ttab = make_ttab();

// ---------------------------------------------------------------------------
// Syndrome -> per-site 6-bit class table (prep).
// _surface_to_flags: s2[n]=surf[(16-n)&15]; cum = inclusive cumsum;
//                    flag[n] = cum[(n+15)&15] & 1
// ---------------------------------------------------------------------------
__device__ __forceinline__ void flags16(const int* surf, unsigned char* f) {
  int cum[16];
  int run = 0;
  for (int n = 0; n < 16; n++) { run += surf[(16 - n) & 15]; cum[n] = run; }
  for (int n = 0; n < 16; n++) f[n] = (unsigned char)(cum[(n + 15) & 15] & 1);
}

__global__ void __launch_bounds__(256)
prep_classes(const int* __restrict__ syndrome, unsigned char* __restrict__ class_tab) {
  __shared__ int sZ0[256], sZ1[256], sZ2[256];
  __shared__ int sX0[16], sX1[16], sX2[16];
  __shared__ unsigned char fz0[256], fz1[256], fz2[256];
  __shared__ unsigned char fx0[16], fx1[16], fx2[16];

  const int b = blockIdx.x;
  const int tid = threadIdx.x;

  sZ0[tid] = 0; sZ1[tid] = 0; sZ2[tid] = 0;
  if (tid < 16) { sX0[tid] = 0; sX1[tid] = 0; sX2[tid] = 0; }
  __syncthreads();

  const int* z = syndrome + (size_t)b * 16384;
  for (int idx = tid; idx < 4096; idx += 256) {
    int d1 = idx >> 8, d2 = (idx >> 4) & 15, d3 = idx & 15;
    int v = z[idx];                       // z_syn
    if (v) {
      atomicAdd(&sZ0[d2 * 16 + d3], v);   // sum over d1
      atomicAdd(&sZ1[d1 * 16 + d3], v);   // sum over d2
      atomicAdd(&sZ2[d1 * 16 + d2], v);   // sum over d3
    }
    int v0 = z[4096 + idx];  if (v0) atomicAdd(&sX0[d3], v0);   // x0: sum over d1,d2
    int v1 = z[8192 + idx];  if (v1) atomicAdd(&sX1[d1], v1);   // x1: sum over d2,d3
    int v2 = z[12288 + idx]; if (v2) atomicAdd(&sX2[d2], v2);   // x2: sum over d1,d3
  }
  __syncthreads();

  if (tid < 48) {
    int which = tid >> 4, row = tid & 15;
    const int* src = (which == 0 ? sZ0 : which == 1 ? sZ1 : sZ2) + row * 16;
    unsigned char* dst = (which == 0 ? fz0 : which == 1 ? fz1 : fz2) + row * 16;
    flags16(src, dst);
  } else if (tid == 48) {
    int tmp[16];                          // x0 rolled by 1 along d3 before flags
    for (int t = 0; t < 16; t++) tmp[t] = sX0[(t + 15) & 15];
    flags16(tmp, fx0);
  } else if (tid == 49) {
    flags16(sX1, fx1);                    // roll was along a summed axis: no-op
  } else if (tid == 50) {
    flags16(sX2, fx2);
  }
  __syncthreads();

  unsigned char* ct = class_tab + (size_t)b * 4096;
  for (int idx = tid; idx < 4096; idx += 256) {
    int i = idx >> 8, j = (idx >> 4) & 15, k = idx & 15;
    int c = fz0[k * 16 + j] | (fx0[k] << 1) | (fz1[i * 16 + k] << 2) |
            (fx1[i] << 3) | (fz2[j * 16 + i] << 4) | (fx2[j] << 5);
    ct[idx] = (unsigned char)c;
  }
}

// ---------------------------------------------------------------------------
// Main streaming kernel: async global->LDS with a 4-deep ring buffer
// (ASYNCcnt path). Each async instruction copies 512 B (2 sites):
// lane t moves bytes [16t, 16t+16). 4-deep keeps 2 KB/wave in flight, i.e.
// ~64 KB per fully occupied WGP -- enough to cover HBM latency at 23.3 TB/s.
// ---------------------------------------------------------------------------
__device__ __forceinline__ void async_copy_b128(unsigned lds_off, unsigned goff,
                                                const float* sbase) {
  // GVS mode: mem = SADDR(64b sgpr pair) + VADDR(32b per-lane) + ioffset
  asm volatile("global_load_async_to_lds_b128 %0, %1, %2"
               :
               : "v"(lds_off), "v"(goff), "s"(sbase)
               : "memory");
}

__global__ void __launch_bounds__(256)
pool_main(const float* __restrict__ x, const unsigned char* __restrict__ class_tab,
          float* __restrict__ out) {
  __shared__ uint32_t s_tt[64 * 64];   // permutation table widened to u32 (16 KB)
  __shared__ float s_xbuf[8][512];     // per-wave ring: 4 bufs x 128 floats (16 KB)

  const int tid = threadIdx.x;
  const unsigned char* tsrc = &g_ttab.t[0][0];
  for (int e = tid; e < 4096; e += 256) s_tt[e] = tsrc[e];
  __syncthreads();

  const int b = blockIdx.x >> 3;     // 128 batch items
  const int slab = blockIdx.x & 7;   // 8 slabs of 512 sites
  const int wave = tid >> 5;         // 8 waves, 64 sites each
  const int lane = tid & 31;

  const int site0 = slab * 512 + wave * 64;
  const float* xb = x + ((size_t)b * 4096 + site0) * 64;
  const unsigned char* cls = class_tab + (size_t)b * 4096 + site0;

  // low 32 bits of the flat address of an LDS object == its LDS byte offset
  const unsigned lds_base = (unsigned)(uintptr_t)&s_xbuf[wave][0];

  float acc0 = 0.f, acc1 = 0.f;

  // prologue: fill ring slots 0..2 (pairs 0..2 of 32)
#pragma unroll
  for (unsigned p = 0; p < 3; ++p)
    async_copy_b128(lds_base + p * 512u + lane * 16u, p * 512u + lane * 16u, xb);

  for (int step = 0; step < 32; ++step) {
    const int nxt = step + 3;                 // next pair to prefetch
    if (nxt < 32) {
      async_copy_b128(lds_base + (unsigned)(nxt & 3) * 512u + lane * 16u,
                      (unsigned)nxt * 512u + lane * 16u, xb);
      asm volatile("s_wait_asynccnt 0x3" ::: "memory");  // oldest fill done
    } else {
      const int rem = 31 - step;              // outstanding younger fills
      if (rem == 2)      asm volatile("s_wait_asynccnt 0x2" ::: "memory");
      else if (rem == 1) asm volatile("s_wait_asynccnt 0x1" ::: "memory");
      else               asm volatile("s_wait_asynccnt 0x0" ::: "memory");
    }
    const float* bufp = &s_xbuf[wave][(step & 3) * 128];
#pragma unroll
    for (int t = 0; t < 2; ++t) {
      int c = cls[2 * step + t];
      uint32_t i0 = s_tt[c * 64 + lane];
      uint32_t i1 = s_tt[c * 64 + lane + 32];
      acc0 += bufp[t * 64 + i0];   // permutation of 64 distinct dwords:
      acc1 += bufp[t * 64 + i1];   // bank-conflict-free LDS gather
    }
  }

  const float scale = 1.f / 4096.f;
  atomicAdd(&out[b * 64 + lane], acc0 * scale);
  atomicAdd(&out[b * 64 + lane + 32], acc1 * scale);
}

__global__ void zero_out(float* out, int n) {
  int i = blockIdx.x * blockDim.x + threadIdx.x;
  if (i < n) out[i] = 0.f;
}

extern "C" void kernel_launch(void* const* d_in, const int* in_sizes, int n_in,
                              void* d_out, int out_size, void* d_ws, size_t ws_size,
                              hipStream_t stream) {
  (void)in_sizes; (void)n_in; (void)out_size; (void)ws_size;
  const float* x = (const float*)d_in[0];
  const int* syn = (const int*)d_in[1];
  float* out = (float*)d_out;
  unsigned char* class_tab = (unsigned char*)d_ws;  // 128*4096 = 512 KB scratch

  zero_out<<<32, 256, 0, stream>>>(out, 128 * 64);
  prep_classes<<<128, 256, 0, stream>>>(syn, class_tab);
  pool_main<<<128 * 8, 256, 0, stream>>>(x, class_tab, out);
}